// AdditiveAttention_62294205662042
// MI455X (gfx1250) — compile-verified
//
#include <hip/hip_runtime.h>
#include <hip/hip_bf16.h>

typedef __attribute__((ext_vector_type(2))) float v2f;
typedef __attribute__((ext_vector_type(8))) float v8f;

// ---------------------------------------------------------------------------
// Generic fp32 GEMM with optional bias using V_WMMA_F32_16X16X4_F32.
//   C[b] = A[b] (MxKd) * B[b] (KdxN) + bias(N)
// Block = 128 threads = 4 waves; block tile = 16(M) x 64(N); each wave owns a
// 16x16 output tile and loops Kd in steps of 4 (one wmma per step).
// WMMA fragment layouts per CDNA5 ISA 7.12.2 (fp32 16x16x4):
//   A: lane l, reg v -> m = l&15, k = 2*(l>>4)+v      (2 VGPRs)
//   B: lane l, reg v -> n = l&15, k = 2*(l>>4)+v      (2 VGPRs)
//   D: reg r, lane l -> m = r + 8*(l>>4), n = l&15    (8 VGPRs)
// ---------------------------------------------------------------------------
__global__ __launch_bounds__(128) void gemm_f32_wmma(
    const float* __restrict__ A, long strideAb,
    const float* __restrict__ B, long strideBb,
    const float* __restrict__ bias,
    float* __restrict__ C, long strideCb,
    int M, int N, int Kd)
{
    const int bz = blockIdx.z;
    A += (long)bz * strideAb;
    B += (long)bz * strideBb;
    C += (long)bz * strideCb;

    const int wave = threadIdx.x >> 5;
    const int lane = threadIdx.x & 31;
    const int mBase = blockIdx.y * 16;
    const int nBase = blockIdx.x * 64 + wave * 16;
    if (mBase >= M || nBase >= N) return;   // wave-uniform: EXEC stays all-1s

    const int l16 = lane & 15;
    const int hk  = (lane >> 4) * 2;        // 0 or 2

    v8f acc = {};
    const float* aPtr = A + (long)(mBase + l16) * Kd + hk;
    const float* bPtr = B + (long)hk * N + nBase + l16;

    for (int k = 0; k < Kd; k += 4) {
        v2f af, bf;
        af.x = aPtr[0];                     // A[m][k+hk]   (contiguous pair)
        af.y = aPtr[1];                     // A[m][k+hk+1]
        bf.x = bPtr[0];                     // B[k+hk][n]
        bf.y = bPtr[(long)N];               // B[k+hk+1][n]
        acc = __builtin_amdgcn_wmma_f32_16x16x4_f32(
            false, af, false, bf, (short)0, acc, false, false);
        aPtr += 4;
        bPtr += (long)4 * N;
    }

    const int nn = nBase + l16;
    const float bb = bias ? bias[nn] : 0.0f;
    const int mTop = mBase + (lane >> 4) * 8;
#pragma unroll
    for (int r = 0; r < 8; ++r) {
        C[(long)(mTop + r) * N + nn] = acc[r] + bb;
    }
}

// ---------------------------------------------------------------------------
// Fused additive-attention score:
//   score[b,q,k] = sum_h wv[h]*tanh(qp[b,q,h] + kp[b,k,h]) + bv
// Never materializes the [B,Q,K,H] tensor. One block (256 thr = 8 waves) per
// (b,q) row; each wave owns one key k at a time; lanes parallelize h (H=256
// -> 8 values/lane, preloaded qp/wv in registers); wave32 xor-shuffle reduce.
// ---------------------------------------------------------------------------
template <int Bc, int Qc, int Kc, int Hc>
__global__ __launch_bounds__(256) void score_kernel(
    const float* __restrict__ qp,   // [B,Q,H]
    const float* __restrict__ kp,   // [B,K,H]
    const float* __restrict__ wv,   // [H]
    const float* __restrict__ bvp,  // [1]
    float* __restrict__ score)      // [B,Q,K]
{
    const int bq   = blockIdx.x;          // 0 .. B*Q-1
    const int b    = bq / Qc;
    const int lane = threadIdx.x & 31;
    const int wave = threadIdx.x >> 5;

    constexpr int J = Hc / 32;            // h-values per lane (8)
    float qv[J], wvv[J];
    const float* qrow = qp + (long)bq * Hc;
#pragma unroll
    for (int j = 0; j < J; ++j) {
        qv[j]  = qrow[lane + 32 * j];
        wvv[j] = wv[lane + 32 * j];
    }
    const float bias = bvp[0];
    const float* kbat = kp + (long)b * Kc * Hc;
    float* srow = score + (long)bq * Kc;

    for (int k = wave; k < Kc; k += 8) {
        const float* krow = kbat + (long)k * Hc;
        float s = 0.0f;
#pragma unroll
        for (int j = 0; j < J; ++j) {
            float t = qv[j] + krow[lane + 32 * j];   // coalesced 128B/load
            s = fmaf(wvv[j], tanhf(t), s);
        }
#pragma unroll
        for (int off = 16; off > 0; off >>= 1)
            s += __shfl_xor(s, off, 32);
        if (lane == 0) srow[k] = s + bias;
    }
}

// ---------------------------------------------------------------------------
// In-place softmax over the last dim (K) of [B*Q, K]. 256 threads per row,
// 2 elements per thread (K=512). Two-pass max/sum with LDS reductions.
// ---------------------------------------------------------------------------
template <int Kc>
__global__ __launch_bounds__(256) void softmax_kernel(float* __restrict__ attn)
{
    __shared__ float redMax[8];
    __shared__ float redSum[8];

    float* p = attn + (long)blockIdx.x * Kc;
    const int t    = threadIdx.x;
    const int lane = t & 31;
    const int wave = t >> 5;

    float x0 = p[t];
    float x1 = p[t + 256];

    float m = fmaxf(x0, x1);
#pragma unroll
    for (int off = 16; off > 0; off >>= 1)
        m = fmaxf(m, __shfl_xor(m, off, 32));
    if (lane == 0) redMax[wave] = m;
    __syncthreads();
    float M = redMax[0];
#pragma unroll
    for (int i = 1; i < 8; ++i) M = fmaxf(M, redMax[i]);

    float e0 = __expf(x0 - M);
    float e1 = __expf(x1 - M);
    float s = e0 + e1;
#pragma unroll
    for (int off = 16; off > 0; off >>= 1)
        s += __shfl_xor(s, off, 32);
    if (lane == 0) redSum[wave] = s;
    __syncthreads();
    float S = redSum[0];
#pragma unroll
    for (int i = 1; i < 8; ++i) S += redSum[i];

    const float inv = 1.0f / S;
    p[t]       = e0 * inv;
    p[t + 256] = e1 * inv;
}

// ---------------------------------------------------------------------------
// Launch: q-proj GEMM, k-proj GEMM, fused score, softmax, context GEMM.
// d_out = [context (B*Q*DV) | attention_weight (B*Q*K)] fp32.
// d_ws  = [qp (B*Q*H) | kp (B*K*H)] fp32 (4 MB).
// ---------------------------------------------------------------------------
extern "C" void kernel_launch(void* const* d_in, const int* in_sizes, int n_in,
                              void* d_out, int out_size, void* d_ws, size_t ws_size,
                              hipStream_t stream) {
    constexpr int B = 4, Q = 512, K = 512, DQ = 512, DK = 512, H = 256, DV = 512;
    (void)in_sizes; (void)n_in; (void)out_size; (void)ws_size;

    const float* query = (const float*)d_in[0];
    const float* key   = (const float*)d_in[1];
    const float* value = (const float*)d_in[2];
    const float* Wq    = (const float*)d_in[3];
    const float* bq    = (const float*)d_in[4];
    const float* Wk    = (const float*)d_in[5];
    const float* bk    = (const float*)d_in[6];
    const float* wv    = (const float*)d_in[7];
    const float* bvp   = (const float*)d_in[8];

    float* ctx  = (float*)d_out;                  // [B,Q,DV]
    float* attn = ctx + (size_t)B * Q * DV;       // [B,Q,K]
    float* qp   = (float*)d_ws;                   // [B,Q,H]
    float* kp   = qp + (size_t)B * Q * H;         // [B,K,H]

    const dim3 blk(128);

    // q = query(2048x512) * Wq(512x256) + bq
    gemm_f32_wmma<<<dim3(H / 64, (B * Q) / 16, 1), blk, 0, stream>>>(
        query, 0L, Wq, 0L, bq, qp, 0L, B * Q, H, DQ);

    // k = key(2048x512) * Wk(512x256) + bk
    gemm_f32_wmma<<<dim3(H / 64, (B * K) / 16, 1), blk, 0, stream>>>(
        key, 0L, Wk, 0L, bk, kp, 0L, B * K, H, DK);

    // score[b,q,k] = sum_h wv[h]*tanh(qp+kp) + bv   (written into attn slot)
    score_kernel<B, Q, K, H><<<B * Q, 256, 0, stream>>>(qp, kp, wv, bvp, attn);

    // softmax over K, in place
    softmax_kernel<K><<<B * Q, 256, 0, stream>>>(attn);

    // context[b] = attn[b](512x512) * value[b](512x512)
    gemm_f32_wmma<<<dim3(DV / 64, Q / 16, B), blk, 0, stream>>>(
        attn, (long)Q * K, value, (long)K * DV, nullptr,
        ctx, (long)Q * DV, Q, DV, K);
}